// SSM_33002528702791
// MI455X (gfx1250) — compile-verified
//
#include <hip/hip_runtime.h>

// ---------------------------------------------------------------------------
// Diagonal SSM  y = causal-conv(u, K),  K(t) = sum_n C_n B_n a_n^t
// Chunked-scan formulation, all matmuls on V_WMMA_F32_16X16X4_F32 (fp32 exact).
//   B=4, D=1024 -> S=4096 sequences, L=2048, N=64 states.
//   Chunk T=16, segment = 16 chunks (256 steps), 8 segments.
// Pass A: state-only scan, dump segment-boundary states (8 MB in d_ws).
// Pass B: 256 tiles x 8 segments = 2048 independent waves, full Y + state.
// Waves are packed 4-per-workgroup (128 threads, one per SIMD32 of a WGP);
// each wave owns its own 16-sequence tile -- no LDS, no barriers.
// ---------------------------------------------------------------------------

typedef __attribute__((ext_vector_type(2))) float v2f;
typedef __attribute__((ext_vector_type(8))) float v8f;

#define SEQS      4096
#define LSEQ      2048
#define NST       64
#define TCH       16      // chunk length (WMMA M/N)
#define NSEG      8
#define CH_SEG    16      // chunks per segment
#define NTILE     256     // 4096 seqs / 16
#define WPB       4       // waves per block

// d_ws float layout
#define OFF_M     0       // 16x16
#define OFF_CM    256     // 16x64
#define OFF_BM    1280    // 64x16
#define OFF_A16   2304    // 64
#define OFF_ST    2560    // states: [256 tiles][8 seg][4 ntile][8 r][32 lane]

__device__ __forceinline__ v8f wmma4_f32(v2f a, v2f b, v8f c) {
  // D = A(16x4) * B(4x16) + C, fp32 throughout.
  return __builtin_amdgcn_wmma_f32_16x16x4_f32(false, a, false, b,
                                               (short)0, c, false, false);
}

// ------------------------- table precompute (tiny) -------------------------
__global__ void ssm_prep(const float* __restrict__ A, const float* __restrict__ Bv,
                         const float* __restrict__ Cv, float* __restrict__ tbl) {
  int t = threadIdx.x;
  float* Mm  = tbl + OFF_M;
  float* Cm  = tbl + OFF_CM;
  float* Bm  = tbl + OFF_BM;
  float* a16 = tbl + OFF_A16;
  if (t < NST) {
    float a = A[t * NST + t];     // diagonal entry
    float b = Bv[t], c = Cv[t];
    float pw[17];
    float p = 1.0f;
    for (int i = 0; i < 17; ++i) { pw[i] = p; p *= a; }
    a16[t] = pw[16];
    for (int k = 0; k < TCH; ++k) Bm[t * TCH + k] = b * pw[15 - k];
    for (int l = 0; l < TCH; ++l) Cm[l * NST + t] = c * pw[l + 1];
  }
  if (t < 256) {                  // M[l][k] = K(l-k) for k<=l else 0
    int l = t >> 4, k = t & 15;
    float v = 0.0f;
    if (k <= l) {
      int d = l - k;
      for (int n = 0; n < NST; ++n) {
        float a = A[n * NST + n];
        float p = 1.0f;
        for (int i = 0; i < d; ++i) p *= a;
        v += Cv[n] * Bv[n] * p;
      }
    }
    Mm[l * TCH + k] = v;
  }
}

// ---------------- pass A: state scan, dump segment boundaries ---------------
__global__ __launch_bounds__(WPB * 32) void ssm_state_pass(
    const float* __restrict__ u, const float* __restrict__ tbl,
    float* __restrict__ states) {
  const int lane = threadIdx.x & 31;
  const int half = lane >> 4;
  const int lm   = lane & 15;
  const int tile = blockIdx.x * WPB + (threadIdx.x >> 5);

  const float* Bm  = tbl + OFF_BM;
  const float* a16 = tbl + OFF_A16;

  v2f bm[4][4];                       // A-operand: Bm rows 16t+lm, k-window 4j
  #pragma unroll
  for (int t = 0; t < 4; ++t)
    #pragma unroll
    for (int j = 0; j < 4; ++j)
      bm[t][j] = *(const v2f*)(Bm + (16 * t + lm) * TCH + 4 * j + 2 * half);

  float av[4][8];                     // a^16 in C/D layout (row = r + 8*half)
  #pragma unroll
  for (int t = 0; t < 4; ++t)
    #pragma unroll
    for (int r = 0; r < 8; ++r)
      av[t][r] = a16[16 * t + r + 8 * half];

  v8f H[4];
  const v8f zero = {0.f, 0.f, 0.f, 0.f, 0.f, 0.f, 0.f, 0.f};
  #pragma unroll
  for (int t = 0; t < 4; ++t) H[t] = zero;

  const float* useq = u + (size_t)(tile * 16 + lm) * LSEQ;

  for (int g = 0; g < NSEG; ++g) {
    // dump state at segment start (straight-line, once per 16 chunks)
    float* sp = states + (size_t)(tile * NSEG + g) * 1024;
    #pragma unroll
    for (int t = 0; t < 4; ++t)
      #pragma unroll
      for (int r = 0; r < 8; ++r)
        sp[t * 256 + r * 32 + lane] = H[t][r];

    const float* useg = useq + g * (CH_SEG * TCH);
    for (int c = 0; c < CH_SEG; ++c) {
      const int cb = c * TCH;
      v2f ub[4];                      // B-operand: U[k][seq], contiguous b64
      #pragma unroll
      for (int j = 0; j < 4; ++j)
        ub[j] = *(const v2f*)(useg + cb + 4 * j + 2 * half);
      __builtin_prefetch((const void*)(useg + cb + 4 * TCH), 0, 0);

      #pragma unroll
      for (int t = 0; t < 4; ++t) {
        #pragma unroll
        for (int r = 0; r < 8; ++r) H[t][r] *= av[t][r];   // diag(a^16) * H
        #pragma unroll
        for (int j = 0; j < 4; ++j) H[t] = wmma4_f32(bm[t][j], ub[j], H[t]);
      }
    }
  }
}

// ---------------- pass B: per-(tile,segment) full computation ---------------
__global__ __launch_bounds__(WPB * 32) void ssm_main(
    const float* __restrict__ u, const float* __restrict__ tbl,
    const float* __restrict__ states, float* __restrict__ y) {
  const int lane = threadIdx.x & 31;
  const int half = lane >> 4;
  const int lm   = lane & 15;
  const int tile = blockIdx.x * WPB + (threadIdx.x >> 5);
  const int seg  = blockIdx.y;

  const float* Mm  = tbl + OFF_M;
  const float* Cm  = tbl + OFF_CM;
  const float* Bm  = tbl + OFF_BM;
  const float* a16 = tbl + OFF_A16;

  v2f mA[4];                                   // triangular-Toeplitz kernel
  #pragma unroll
  for (int j = 0; j < 4; ++j)
    mA[j] = *(const v2f*)(Mm + lm * TCH + 4 * j + 2 * half);

  v2f cmA[4][4];                               // Cm rows lm, n-window 16t+4j
  #pragma unroll
  for (int t = 0; t < 4; ++t)
    #pragma unroll
    for (int j = 0; j < 4; ++j)
      cmA[t][j] = *(const v2f*)(Cm + lm * NST + 16 * t + 4 * j + 2 * half);

  v2f bm[4][4];
  #pragma unroll
  for (int t = 0; t < 4; ++t)
    #pragma unroll
    for (int j = 0; j < 4; ++j)
      bm[t][j] = *(const v2f*)(Bm + (16 * t + lm) * TCH + 4 * j + 2 * half);

  float av[4][8];
  #pragma unroll
  for (int t = 0; t < 4; ++t)
    #pragma unroll
    for (int r = 0; r < 8; ++r)
      av[t][r] = a16[16 * t + r + 8 * half];

  v8f H[4];                                    // segment-start state (C/D layout)
  {
    const float* sp = states + (size_t)(tile * NSEG + seg) * 1024;
    #pragma unroll
    for (int t = 0; t < 4; ++t)
      #pragma unroll
      for (int r = 0; r < 8; ++r)
        H[t][r] = sp[t * 256 + r * 32 + lane];
  }

  const float* useq = u + (size_t)(tile * 16 + lm) * LSEQ + seg * (CH_SEG * TCH);
  float*       yseq = y + (size_t)(tile * 16 + lm) * LSEQ + seg * (CH_SEG * TCH);
  const v8f zero = {0.f, 0.f, 0.f, 0.f, 0.f, 0.f, 0.f, 0.f};

  for (int c = 0; c < CH_SEG; ++c) {
    const int cb = c * TCH;
    v2f ub[4];
    #pragma unroll
    for (int j = 0; j < 4; ++j)
      ub[j] = *(const v2f*)(useq + cb + 4 * j + 2 * half);
    __builtin_prefetch((const void*)(useq + cb + 4 * TCH), 0, 0);

    // Y = M * U  (intra-chunk triangular convolution)
    v8f yacc = zero;
    #pragma unroll
    for (int j = 0; j < 4; ++j) yacc = wmma4_f32(mA[j], ub[j], yacc);

    // Y += Cm * H : convert H from C/D layout to B-operand layout.
    // Lanes hold the same column in both layouts; only a half-wave swap
    // (shfl_xor 16) of the right register pair is needed per k-step.
    #pragma unroll
    for (int t = 0; t < 4; ++t) {
      #pragma unroll
      for (int j = 0; j < 4; ++j) {
        const int jj = j & 1;
        float e0 = H[t][4 * jj + 0], e1 = H[t][4 * jj + 1];
        float e2 = H[t][4 * jj + 2], e3 = H[t][4 * jj + 3];
        float sx = half ? e0 : e2;           // what my partner needs
        float sy = half ? e1 : e3;
        float px = __shfl_xor(sx, 16);
        float py = __shfl_xor(sy, 16);
        float ox = half ? e2 : e0;           // what I own
        float oy = half ? e3 : e1;
        bool useOwn = (half == (j >> 1));
        v2f hb;
        hb.x = useOwn ? ox : px;
        hb.y = useOwn ? oy : py;
        yacc = wmma4_f32(cmA[t][j], hb, yacc);
      }
    }

    // H' = diag(a^16) H + Bm * U   (not needed after last chunk; uniform branch)
    if (c != CH_SEG - 1) {
      #pragma unroll
      for (int t = 0; t < 4; ++t) {
        #pragma unroll
        for (int r = 0; r < 8; ++r) H[t][r] *= av[t][r];
        #pragma unroll
        for (int j = 0; j < 4; ++j) H[t] = wmma4_f32(bm[t][j], ub[j], H[t]);
      }
    }

    // store Y: lane holds times r+8*half of sequence lm (contiguous per row)
    #pragma unroll
    for (int r = 0; r < 8; ++r)
      yseq[cb + r + 8 * half] = yacc[r];
  }
}

// ------------------------------- launcher ----------------------------------
extern "C" void kernel_launch(void* const* d_in, const int* in_sizes, int n_in,
                              void* d_out, int out_size, void* d_ws, size_t ws_size,
                              hipStream_t stream) {
  const float* u  = (const float*)d_in[0];   // (4,1024,2048) f32
  const float* A  = (const float*)d_in[1];   // (64,64) f32 (diagonal)
  const float* Bv = (const float*)d_in[2];   // (64,) f32
  const float* Cv = (const float*)d_in[3];   // (64,) f32
  float* tbl    = (float*)d_ws;              // tables + 8 MB of states
  float* states = tbl + OFF_ST;
  float* y      = (float*)d_out;             // (4,1024,2048) f32

  ssm_prep<<<dim3(1), dim3(256), 0, stream>>>(A, Bv, Cv, tbl);
  ssm_state_pass<<<dim3(NTILE / WPB), dim3(WPB * 32), 0, stream>>>(u, tbl, states);
  ssm_main<<<dim3(NTILE / WPB, NSEG), dim3(WPB * 32), 0, stream>>>(u, tbl, states, y);
}